// Global_Attention_65017214927344
// MI455X (gfx1250) — compile-verified
//
#include <hip/hip_runtime.h>
#include <hip/hip_bf16.h>

typedef __attribute__((ext_vector_type(2))) float v2f;
typedef __attribute__((ext_vector_type(8))) float v8f;

#define QSCALE 0.17677669529663687f /* 32^-0.5 */

// ---------------------------------------------------------------------------
// Small layout kernels
// ---------------------------------------------------------------------------

// dst[k*O + o] = src[o*K + k]   (weight transpose to K-major)
__global__ void transpose_k(const float* __restrict__ src, float* __restrict__ dst,
                            int O, int K, int total) {
    int idx = blockIdx.x * blockDim.x + threadIdx.x;
    if (idx >= total) return;
    int o = idx % O;
    int k = idx / O;
    dst[idx] = src[(size_t)o * K + k];
}

// pool_w [256,256,8,8] -> poolT[(q*256+c)*256 + o] = pool_w[o*16384 + c*64 + q]
__global__ void pool_transpose(const float* __restrict__ src, float* __restrict__ dst,
                               int total) {
    int idx = blockIdx.x * blockDim.x + threadIdx.x;
    if (idx >= total) return;
    int o = idx % 256;
    int k = idx / 256;
    int c = k % 256;
    int q = k / 256;
    dst[idx] = src[(size_t)o * 16384 + c * 64 + q];
}

// window partition: xw[(n*64+q)*256 + c] = x[bi, c, wh*8+i, ww*8+j]
__global__ void gather_xw(const float* __restrict__ x, float* __restrict__ xw, int total) {
    int idx = blockIdx.x * blockDim.x + threadIdx.x;
    if (idx >= total) return;
    int c = idx % 256;
    int t = idx / 256;
    int q = t % 64;
    int n = t / 64;
    int bi = n / 144;
    int l  = n % 144;
    int wh = l / 12, ww = l % 12;
    int i = q / 8, j = q % 8;
    xw[idx] = x[(((size_t)bi * 256 + c) * 96 + wh * 8 + i) * 96 + ww * 8 + j];
}

// ---------------------------------------------------------------------------
// Generic fp32 WMMA GEMM: C[M,N] = scale * (A[M,K] @ B[K,N]) + bias[N]
// A row-major, B K-major ([K][N]), all dims multiples of 16 (K mult of 4),
// M % 32 == 0, N % 64 == 0. 8 waves/block, one 16x16 tile per wave.
// ---------------------------------------------------------------------------
__global__ __launch_bounds__(256) void gemm_wmma_f32(
    const float* __restrict__ A, const float* __restrict__ B,
    float* __restrict__ C, const float* __restrict__ bias,
    int M, int N, int K, float scale)
{
    int wave = threadIdx.x >> 5;
    int lane = threadIdx.x & 31;
    int hf   = lane >> 4;     // half-wave select (K pair 0/1 vs 2/3)
    int r    = lane & 15;
    int m0 = blockIdx.y * 32 + (wave >> 2) * 16;
    int n0 = blockIdx.x * 64 + (wave & 3) * 16;

    const float* Arow = A + (size_t)(m0 + r) * K + 2 * hf;      // A[m0+r, k+2*hf]
    const float* Bcol = B + (size_t)(2 * hf) * N + n0 + r;      // B[k+2*hf, n0+r]

    v8f acc = {};
    #pragma unroll 8
    for (int k = 0; k < K; k += 4) {
        v2f a, b;
        a.x = Arow[k];
        a.y = Arow[k + 1];
        b.x = Bcol[(size_t)k * N];
        b.y = Bcol[(size_t)k * N + N];
        acc = __builtin_amdgcn_wmma_f32_16x16x4_f32(false, a, false, b,
                                                    (short)0, acc, false, false);
    }
    int col = n0 + r;
    float bv = bias ? bias[col] : 0.0f;
    #pragma unroll
    for (int vg = 0; vg < 8; ++vg)
        C[(size_t)(m0 + vg + 8 * hf) * N + col] = acc[vg] * scale + bv;
}

// ---------------------------------------------------------------------------
// Attention: one block per (g, bb, h). q rows n=g*8+bb, KV of batch bb (the
// faithful n%B pairing). attn = softmax(q@k^T) @ v, written in-place into the
// q buffer region (disjoint per block; reads complete before writes).
// ---------------------------------------------------------------------------
#define LDST 145
__global__ __launch_bounds__(128) void attn_kernel(
    float* __restrict__ qmat, const float* __restrict__ kv)
{
    __shared__ float attn[64 * LDST];
    int id = blockIdx.x;
    int h  = id & 7;
    int bb = (id >> 3) & 7;
    int g  = id >> 6;
    int n  = g * 8 + bb;

    int wave = threadIdx.x >> 5;
    int lane = threadIdx.x & 31;
    int hf   = lane >> 4;
    int r    = lane & 15;
    int m0   = wave * 16;

    const float* qbase = qmat + ((size_t)n * 64 + m0 + r) * 256 + h * 32;
    const float* kvb   = kv + (size_t)bb * 144 * 512 + h * 32;      // k part
    const float* vb    = kvb + 256;                                  // v part

    // --- GEMM1: attn[64,144] = q[64,32] @ k^T[32,144] ---
    for (int nt = 0; nt < 9; ++nt) {
        int n0 = nt * 16;
        const float* kcol = kvb + (size_t)(n0 + r) * 512 + 2 * hf;
        v8f acc = {};
        #pragma unroll
        for (int k = 0; k < 32; k += 4) {
            v2f a, b;
            a.x = qbase[k + 2 * hf];
            a.y = qbase[k + 2 * hf + 1];
            b.x = kcol[k];
            b.y = kcol[k + 1];
            acc = __builtin_amdgcn_wmma_f32_16x16x4_f32(false, a, false, b,
                                                        (short)0, acc, false, false);
        }
        #pragma unroll
        for (int vg = 0; vg < 8; ++vg)
            attn[(m0 + vg + 8 * hf) * LDST + n0 + r] = acc[vg];
    }
    __syncthreads();

    // --- softmax over 144 keys, one thread per query row ---
    if (threadIdx.x < 64) {
        float* row = attn + threadIdx.x * LDST;
        float mx = row[0];
        for (int k = 1; k < 144; ++k) mx = fmaxf(mx, row[k]);
        float s = 0.0f;
        for (int k = 0; k < 144; ++k) { float e = __expf(row[k] - mx); row[k] = e; s += e; }
        float inv = 1.0f / s;
        for (int k = 0; k < 144; ++k) row[k] *= inv;
    }
    __syncthreads();

    // --- GEMM2: out[64,32] = attn[64,144] @ v[144,32] ---
    const float* arow = attn + (m0 + r) * LDST + 2 * hf;
    for (int nt = 0; nt < 2; ++nt) {
        int n0 = nt * 16;
        v8f acc = {};
        #pragma unroll 4
        for (int k = 0; k < 144; k += 4) {
            v2f a, b;
            a.x = arow[k];
            a.y = arow[k + 1];
            b.x = vb[(size_t)(k + 2 * hf) * 512 + n0 + r];
            b.y = vb[(size_t)(k + 2 * hf + 1) * 512 + n0 + r];
            acc = __builtin_amdgcn_wmma_f32_16x16x4_f32(false, a, false, b,
                                                        (short)0, acc, false, false);
        }
        #pragma unroll
        for (int vg = 0; vg < 8; ++vg)
            qmat[((size_t)n * 64 + m0 + vg + 8 * hf) * 256 + h * 32 + n0 + r] = acc[vg];
    }
}

// ---------------------------------------------------------------------------
// Depthwise 3x3 RPE (SAME, cross-correlation) + bias, accumulated into F
// (raw-reshape semantics: flat index spaces coincide).
// ---------------------------------------------------------------------------
__global__ void rpe_add_kernel(const float* __restrict__ x, const float* __restrict__ rw,
                               const float* __restrict__ rb, float* __restrict__ F,
                               int total) {
    int idx = blockIdx.x * blockDim.x + threadIdx.x;
    if (idx >= total) return;
    int xx = idx % 96;
    int t  = idx / 96;
    int y  = t % 96; t /= 96;
    int c  = t % 256;
    int bi = t / 256;
    const float* xp = x + (size_t)(bi * 256 + c) * 9216;
    const float* wp = rw + c * 9;
    float acc = rb[c];
    #pragma unroll
    for (int dy = -1; dy <= 1; ++dy) {
        int yy = y + dy;
        if (yy < 0 || yy >= 96) continue;
        #pragma unroll
        for (int dx = -1; dx <= 1; ++dx) {
            int xc = xx + dx;
            if (xc < 0 || xc >= 96) continue;
            acc += xp[yy * 96 + xc] * wp[(dy + 1) * 3 + (dx + 1)];
        }
    }
    F[idx] += acc;
}

// ---------------------------------------------------------------------------
extern "C" void kernel_launch(void* const* d_in, const int* in_sizes, int n_in,
                              void* d_out, int out_size, void* d_ws, size_t ws_size,
                              hipStream_t stream) {
    const float* x      = (const float*)d_in[0];
    const float* Wq     = (const float*)d_in[1];
    const float* Wkv    = (const float*)d_in[2];
    const float* pool_w = (const float*)d_in[3];
    const float* pool_b = (const float*)d_in[4];
    const float* rpe_w  = (const float*)d_in[5];
    const float* rpe_b  = (const float*)d_in[6];
    const float* proj_w = (const float*)d_in[7];
    const float* proj_b = (const float*)d_in[8];
    float* out = (float*)d_out;

    // workspace layout (floats)
    float* ws    = (float*)d_ws;
    float* xw    = ws;                   // 73728 x 256  (== 1152 x 16384)
    float* qmat  = xw + 18874368;        // 73728 x 256  (Q proj, then attn out / F)
    float* poolT = qmat + 18874368;      // 16384 x 256
    float* WqT   = poolT + 4194304;      // 256 x 256
    float* WkvT  = WqT + 65536;          // 256 x 512
    float* projT = WkvT + 131072;        // 256 x 256
    float* poolx = projT + 65536;        // 1152 x 256
    float* kvb   = poolx + 294912;       // 1152 x 512
    (void)ws_size; (void)in_sizes; (void)n_in; (void)out_size;

    const int TB = 256;
    // 1) weight layout prep
    transpose_k<<<(65536 + TB - 1) / TB, TB, 0, stream>>>(Wq, WqT, 256, 256, 65536);
    transpose_k<<<(131072 + TB - 1) / TB, TB, 0, stream>>>(Wkv, WkvT, 512, 256, 131072);
    transpose_k<<<(65536 + TB - 1) / TB, TB, 0, stream>>>(proj_w, projT, 256, 256, 65536);
    pool_transpose<<<(4194304 + TB - 1) / TB, TB, 0, stream>>>(pool_w, poolT, 4194304);

    // 2) window partition
    gather_xw<<<(18874368 + TB - 1) / TB, TB, 0, stream>>>(x, xw, 18874368);

    // 3) pooling GEMM: pool_x[1152,256] = xw[1152,16384] @ poolT + pool_b
    gemm_wmma_f32<<<dim3(256 / 64, 1152 / 32), TB, 0, stream>>>(
        xw, poolT, poolx, pool_b, 1152, 256, 16384, 1.0f);

    // 4) Q projection (scale folded): qmat[73728,256] = SCALE * xw @ WqT
    gemm_wmma_f32<<<dim3(256 / 64, 73728 / 32), TB, 0, stream>>>(
        xw, WqT, qmat, nullptr, 73728, 256, 256, QSCALE);

    // 5) KV projection: kv[1152,512] = pool_x @ WkvT
    gemm_wmma_f32<<<dim3(512 / 64, 1152 / 32), TB, 0, stream>>>(
        poolx, WkvT, kvb, nullptr, 1152, 512, 256, 1.0f);

    // 6) attention per (group, batch, head) -> in-place into qmat (= F)
    attn_kernel<<<144 * 8 * 8, 128, 0, stream>>>(qmat, kvb);

    // 7) depthwise RPE + bias accumulated into F
    rpe_add_kernel<<<(18874368 + TB - 1) / TB, TB, 0, stream>>>(
        x, rpe_w, rpe_b, qmat, 18874368);

    // 8) output projection: y[73728,256] = F @ projT + proj_b
    gemm_wmma_f32<<<dim3(256 / 64, 73728 / 32), TB, 0, stream>>>(
        qmat, projT, out, proj_b, 73728, 256, 256, 1.0f);
}